// Net_88089779241116
// MI455X (gfx1250) — compile-verified
//
#include <hip/hip_runtime.h>
#include <math.h>

// ---------------- constants ----------------
#define KBINS 25          // 5x5 spline kernel bins
#define MDIM  5

typedef __attribute__((ext_vector_type(2))) float v2f;
typedef __attribute__((ext_vector_type(8))) float v8f;

// ---------------- device helpers ----------------
static __device__ __forceinline__ void atomicMaxF(float* addr, float val) {
  // IEEE ordering trick; valid for mixed signs, init -inf (or 0 for >=0 data)
  if (val >= 0.0f) atomicMax((int*)addr, __float_as_int(val));
  else             atomicMin((unsigned int*)addr, __float_as_uint(val));
}

// ---------------- kernels ----------------

// global max over all edges/components of |pos[src]-pos[dst]|
__global__ void k_edge_absmax(const float* __restrict__ pos,
                              const int* __restrict__ ei, int E,
                              float* __restrict__ maxd) {
  int t = blockIdx.x * blockDim.x + threadIdx.x;
  float m = 0.0f;
  if (t < E) {
    int dst = ei[t], src = ei[E + t];
    float dx = fabsf(pos[2 * src]     - pos[2 * dst]);
    float dy = fabsf(pos[2 * src + 1] - pos[2 * dst + 1]);
    m = fmaxf(dx, dy);
  }
  #pragma unroll
  for (int o = 16; o > 0; o >>= 1) m = fmaxf(m, __shfl_xor(m, o, 32));
  if ((threadIdx.x & 31) == 0) atomicMaxF(maxd, m);   // maxd pre-zeroed, m>=0
}

// one thread per (edge, channel): 4-tap bilinear spline scatter-add + degree.
// acc has row stride ldA (>= KBINS*Cin, padded to a multiple of 4).
__global__ void k_spline_scatter(const float* __restrict__ x,
                                 const float* __restrict__ pos,
                                 const int* __restrict__ ei,
                                 const float* __restrict__ maxd,
                                 float* __restrict__ acc,
                                 float* __restrict__ deg,
                                 int E, int Cin, int ldA) {
  long long t = (long long)blockIdx.x * blockDim.x + threadIdx.x;
  long long total = (long long)E * Cin;
  if (t >= total) return;
  int e = (int)(t / Cin);
  int c = (int)(t % Cin);
  int dst = ei[e], src = ei[E + e];
  float inv = 1.0f / (2.0f * maxd[0]);
  float px = (pos[2 * src]     - pos[2 * dst])     * inv + 0.5f;
  float py = (pos[2 * src + 1] - pos[2 * dst + 1]) * inv + 0.5f;
  float ppx = px * (float)(MDIM - 1);
  float ppy = py * (float)(MDIM - 1);
  float k0x = fminf(fmaxf(floorf(ppx), 0.0f), (float)(MDIM - 2));
  float k0y = fminf(fmaxf(floorf(ppy), 0.0f), (float)(MDIM - 2));
  float fx = ppx - k0x, fy = ppy - k0y;
  int ix = (int)k0x, iy = (int)k0y;
  float wx[2] = {1.0f - fx, fx};
  float wy[2] = {1.0f - fy, fy};
  float xv = x[(long long)src * Cin + c];
  long long base = (long long)dst * ldA + c;
  #pragma unroll
  for (int a = 0; a < 2; ++a)
    #pragma unroll
    for (int b = 0; b < 2; ++b) {
      int widx = (ix + a) * MDIM + (iy + b);
      atomicAdd(&acc[base + (long long)widx * Cin], wx[a] * wy[b] * xv);
    }
  if (c == 0) atomicAdd(&deg[dst], 1.0f);
}

// zero-padded weight copy: Wp[k,n] = (k<Kd) ? W[k,n] : 0, for k < KdP
__global__ void k_pad_w(const float* __restrict__ W, float* __restrict__ Wp,
                        int Kd, int KdP, int N) {
  int t = blockIdx.x * blockDim.x + threadIdx.x;
  if (t >= KdP * N) return;
  Wp[t] = ((t / N) < Kd) ? W[t] : 0.0f;
}

// C[MxN] = A[MxKdP](stride lda) * Bm[KdPxN], row-major.
// One wave computes a 16 x (16*NT) strip. Software-pipelined:
// fragments for step k+1 are loaded before the WMMAs of step k, so the
// compiler can issue partial s_wait_loadcnt and overlap loads with
// V_WMMA_F32_16X16X4_F32 issue. KdP % 4 == 0, lda even -> branch-free,
// 8B-aligned single-b64 A loads.
template <int NT>
__global__ void k_gemm_wmma(const float* __restrict__ A, int lda,
                            const float* __restrict__ Bm,
                            float* __restrict__ C,
                            int M, int KdP, int N) {
  int wave = blockIdx.x * (blockDim.x >> 5) + (threadIdx.x >> 5);
  int lane = threadIdx.x & 31;
  const int TN = 16 * NT;
  int groupsN = N / TN;
  int nTiles = (M >> 4) * groupsN;
  if (wave >= nTiles) return;
  int mt = wave / groupsN;
  int ng = wave - mt * groupsN;
  int ln = lane & 15;
  int khalf = (lane >> 4) << 1;              // 0 (lanes 0-15) / 2 (lanes 16-31)
  int row  = (mt << 4) + ln;
  int col0 = ng * TN + ln;
  const float* ap = A + (long long)row * lda + khalf;
  const float* bp = Bm + (long long)khalf * N + col0;
  v8f zero = {};
  v8f acc[NT];
  #pragma unroll
  for (int j = 0; j < NT; ++j) acc[j] = zero;

  // prologue: fragments for k = 0
  v2f a_cur = *(const v2f*)ap;               // one global_load_b64
  v2f b_cur[NT];
  #pragma unroll
  for (int j = 0; j < NT; ++j) {
    b_cur[j].x = bp[j * 16];
    b_cur[j].y = bp[j * 16 + N];
  }

  #pragma unroll 2
  for (int k = 4; k < KdP; k += 4) {
    ap += 4;
    bp += (long long)4 * N;
    v2f a_nxt = *(const v2f*)ap;             // issue next loads first
    v2f b_nxt[NT];
    #pragma unroll
    for (int j = 0; j < NT; ++j) {
      b_nxt[j].x = bp[j * 16];
      b_nxt[j].y = bp[j * 16 + N];
    }
    __builtin_prefetch((const void*)(ap + 32), 0, 3);  // WGP-scope prefetch
    #pragma unroll
    for (int j = 0; j < NT; ++j)
      acc[j] = __builtin_amdgcn_wmma_f32_16x16x4_f32(
          /*neg_a=*/false, a_cur, /*neg_b=*/false, b_cur[j],
          /*c_mod=*/(short)0, acc[j], /*reuse_a=*/false, /*reuse_b=*/false);
    a_cur = a_nxt;
    #pragma unroll
    for (int j = 0; j < NT; ++j) b_cur[j] = b_nxt[j];
  }
  // epilogue: last step
  #pragma unroll
  for (int j = 0; j < NT; ++j)
    acc[j] = __builtin_amdgcn_wmma_f32_16x16x4_f32(
        false, a_cur, false, b_cur[j], (short)0, acc[j], false, false);

  int rbase = (mt << 4) + ((lane >> 4) << 3);   // lanes 16-31 hold M=8..15
  float* cp = C + (long long)rbase * N + col0;
  #pragma unroll
  for (int j = 0; j < NT; ++j)
    #pragma unroll
    for (int v = 0; v < 8; ++v)
      cp[(long long)v * N + j * 16] = acc[j][v];
}

// y[n,o] = elu( y[n,o]/max(deg[n],1) + sum_i x[n,i]*R[i,o] )   (in place)
__global__ void k_conv_epilogue(float* __restrict__ y,
                                const float* __restrict__ x,
                                const float* __restrict__ R,
                                const float* __restrict__ deg,
                                int Nn, int Cin, int Cout) {
  long long t = (long long)blockIdx.x * blockDim.x + threadIdx.x;
  long long total = (long long)Nn * Cout;
  if (t >= total) return;
  int n = (int)(t / Cout);
  int o = (int)(t % Cout);
  float s = y[t] / fmaxf(deg[n], 1.0f);
  float r = 0.0f;
  const float* xr = x + (long long)n * Cin;
  for (int i = 0; i < Cin; ++i) r += xr[i] * R[i * Cout + o];
  float v = s + r;
  y[t] = (v > 0.0f) ? v : expm1f(v);
}

__global__ void k_fill_neginf(float* __restrict__ p, long long n) {
  long long t = (long long)blockIdx.x * blockDim.x + threadIdx.x;
  if (t < n) p[t] = -INFINITY;
}

__global__ void k_pool_scatter(const float* __restrict__ h,
                               const int* __restrict__ cl,
                               float* __restrict__ p, int Nn, int C) {
  long long t = (long long)blockIdx.x * blockDim.x + threadIdx.x;
  long long total = (long long)Nn * C;
  if (t >= total) return;
  int n = (int)(t / C);
  int c = (int)(t % C);
  atomicMaxF(&p[(long long)cl[n] * C + c], h[t]);
}

__global__ void k_pool_fixup(float* __restrict__ p, long long n) {
  long long t = (long long)blockIdx.x * blockDim.x + threadIdx.x;
  if (t < n && p[t] == -INFINITY) p[t] = 0.0f;
}

__global__ void k_bias_elu(float* __restrict__ y, const float* __restrict__ b,
                           long long total, int N) {
  long long t = (long long)blockIdx.x * blockDim.x + threadIdx.x;
  if (t >= total) return;
  float v = y[t] + b[(int)(t % N)];
  y[t] = (v > 0.0f) ? v : expm1f(v);
}

// logits = h(1024x128) @ w(128x10) + b ; out = log_softmax(logits)
__global__ void k_fc2_logsoftmax(const float* __restrict__ h,
                                 const float* __restrict__ w,
                                 const float* __restrict__ b,
                                 float* __restrict__ out, int Bn) {
  int r = blockIdx.x * blockDim.x + threadIdx.x;
  if (r >= Bn) return;
  float logits[10];
  const float* hr = h + (long long)r * 128;
  #pragma unroll
  for (int o = 0; o < 10; ++o) {
    float s = b[o];
    for (int i = 0; i < 128; ++i) s += hr[i] * w[i * 10 + o];
    logits[o] = s;
  }
  float mx = logits[0];
  #pragma unroll
  for (int o = 1; o < 10; ++o) mx = fmaxf(mx, logits[o]);
  float se = 0.0f;
  #pragma unroll
  for (int o = 0; o < 10; ++o) se += expf(logits[o] - mx);
  float lse = mx + logf(se);
  #pragma unroll
  for (int o = 0; o < 10; ++o) out[(long long)r * 10 + o] = logits[o] - lse;
}

// ---------------- host orchestration ----------------
static inline int cdiv_ll(long long a, int b) { return (int)((a + b - 1) / b); }

static void launch_gemm(const float* A, int lda, const float* Bm, float* C,
                        int M, int KdP, int N, hipStream_t stream) {
  if (N % 64 == 0) {
    long long waves = (long long)(M >> 4) * (N / 64);
    k_gemm_wmma<4><<<cdiv_ll(waves * 32, 128), 128, 0, stream>>>(A, lda, Bm, C,
                                                                 M, KdP, N);
  } else {  // N == 32
    long long waves = (long long)(M >> 4) * (N / 32);
    k_gemm_wmma<2><<<cdiv_ll(waves * 32, 128), 128, 0, stream>>>(A, lda, Bm, C,
                                                                 M, KdP, N);
  }
}

static void run_conv(const float* x, const float* pos, const int* ei,
                     const float* Bmat, const float* R,
                     int Nn, int E, int Cin, int Cout, int KdP,
                     float* acc, float* deg, float* maxd, float* h,
                     hipStream_t stream) {
  hipMemsetAsync(acc,  0, (size_t)Nn * KdP * sizeof(float), stream);
  hipMemsetAsync(deg,  0, (size_t)Nn * sizeof(float), stream);
  hipMemsetAsync(maxd, 0, sizeof(float), stream);
  const int blk = 256;
  k_edge_absmax<<<cdiv_ll(E, blk), blk, 0, stream>>>(pos, ei, E, maxd);
  long long tot = (long long)E * Cin;
  k_spline_scatter<<<cdiv_ll(tot, blk), blk, 0, stream>>>(x, pos, ei, maxd,
                                                          acc, deg, E, Cin,
                                                          KdP);
  launch_gemm(acc, KdP, Bmat, h, Nn, KdP, Cout, stream);
  long long ne = (long long)Nn * Cout;
  k_conv_epilogue<<<cdiv_ll(ne, blk), blk, 0, stream>>>(h, x, R, deg, Nn, Cin,
                                                        Cout);
}

static void run_pool(const float* h, const int* cl, float* p,
                     int Nn, int Nout, int C, hipStream_t stream) {
  const int blk = 256;
  long long np = (long long)Nout * C;
  long long nh = (long long)Nn * C;
  k_fill_neginf<<<cdiv_ll(np, blk), blk, 0, stream>>>(p, np);
  k_pool_scatter<<<cdiv_ll(nh, blk), blk, 0, stream>>>(h, cl, p, Nn, C);
  k_pool_fixup<<<cdiv_ll(np, blk), blk, 0, stream>>>(p, np);
}

extern "C" void kernel_launch(void* const* d_in, const int* in_sizes, int n_in,
                              void* d_out, int out_size, void* d_ws,
                              size_t ws_size, hipStream_t stream) {
  (void)in_sizes; (void)n_in; (void)out_size; (void)ws_size;
  const int Bv = 1024;
  const int N1 = Bv * 784, E1 = N1 * 8;
  const int N2 = Bv * 36,  E2 = N2 * 8;
  const int N3 = Bv * 16,  E3 = N3 * 8;
  const int NC = Bv * 4;

  const float* x     = (const float*)d_in[0];
  const float* pos1  = (const float*)d_in[1];
  const float* pos2  = (const float*)d_in[2];
  const float* pos3  = (const float*)d_in[3];
  const float* W1    = (const float*)d_in[4];
  const float* R1    = (const float*)d_in[5];
  const float* W2    = (const float*)d_in[6];
  const float* R2    = (const float*)d_in[7];
  const float* W3    = (const float*)d_in[8];
  const float* R3    = (const float*)d_in[9];
  const float* fc1w  = (const float*)d_in[10];
  const float* fc1b  = (const float*)d_in[11];
  const float* fc2w  = (const float*)d_in[12];
  const float* fc2b  = (const float*)d_in[13];
  const int*   ei1   = (const int*)d_in[14];
  const int*   ei2   = (const int*)d_in[15];
  const int*   ei3   = (const int*)d_in[16];
  const int*   cl1   = (const int*)d_in[17];
  const int*   cl2   = (const int*)d_in[18];
  const int*   cl3   = (const int*)d_in[19];
  float* out = (float*)d_out;

  // workspace layout (floats); peak ~235 MB
  float* ws = (float*)d_ws;
  float* accr  = ws;                    // max(acc) = N2*800 = 29,491,200
  float* hr    = ws + 29491200LL;       // max(h)   = N1*32  = 25,690,112
  float* degr  = ws + 55181312LL;       // max(deg) = N1     =    802,816
  float* p1    = ws + 55984128LL;       // N2*32 = 1,179,648
  float* p2    = ws + 57163776LL;       // N3*64 = 1,048,576
  float* p3    = ws + 58212352LL;       // NC*64 =   262,144  (== B x 256)
  float* fc1o  = ws + 58474496LL;       // B*128 =   131,072
  float* maxd  = ws + 58605568LL;       // 1 (+pad)
  float* w1pad = ws + 58605572LL;       // 28*32 = 896

  // layer 1: conv(1->32), Kd=25 padded to 28 (acc stride + zero-padded W1)
  k_pad_w<<<cdiv_ll(28 * 32, 256), 256, 0, stream>>>(W1, w1pad, 25, 28, 32);
  run_conv(x, pos1, ei1, w1pad, R1, N1, E1, 1, 32, 28, accr, degr, maxd, hr,
           stream);
  run_pool(hr, cl1, p1, N1, N2, 32, stream);

  // layer 2: conv(32->64), Kd=800 (already 4-aligned)
  run_conv(p1, pos2, ei2, W2, R2, N2, E2, 32, 64, 800, accr, degr, maxd, hr,
           stream);
  run_pool(hr, cl2, p2, N2, N3, 64, stream);

  // layer 3: conv(64->64), Kd=1600
  run_conv(p2, pos3, ei3, W3, R3, N3, E3, 64, 64, 1600, accr, degr, maxd, hr,
           stream);
  run_pool(hr, cl3, p3, N3, NC, 64, stream);

  // fc1: (B,256) @ (256,128) via WMMA, + bias + elu
  launch_gemm(p3, 256, fc1w, fc1o, Bv, 256, 128, stream);
  long long t1 = (long long)Bv * 128;
  k_bias_elu<<<cdiv_ll(t1, 256), 256, 0, stream>>>(fc1o, fc1b, t1, 128);

  // fc2 + log_softmax -> d_out (B,10)
  k_fc2_logsoftmax<<<cdiv_ll(Bv, 256), 256, 0, stream>>>(fc1o, fc2w, fc2b, out,
                                                         Bv);
}